// QuanvolutionHybrid_65481071397707
// MI455X (gfx1250) — compile-verified
//
#include <hip/hip_runtime.h>
#include <math.h>

typedef float v2f __attribute__((ext_vector_type(2)));
typedef float v8f __attribute__((ext_vector_type(8)));

#define NW      4
#define PATCHES 196
#define PADP    208
#define HID     64
#define NCLS    10
#define IMG     784
#define HSTRIDE 65   // 64 + 1 pad -> conflict-free column reads

// ---------------------------------------------------------------------------
// Kernel 1: one block per image. Quantum sim -> WMMA f32 16x16x4 for
// meas@w1^T (+bias, ReLU) -> scores -> softmax -> attention-weighted flat.
// Writes flat activations to workspace (or computes logits in-block fallback).
// ---------------------------------------------------------------------------
__global__ __launch_bounds__(256) void quanv_fused(
    const float* __restrict__ x,  const float* __restrict__ vparams,
    const float* __restrict__ w1, const float* __restrict__ b1,
    const float* __restrict__ w2, const float* __restrict__ b2,
    const float* __restrict__ Wf, const float* __restrict__ bf,
    float* __restrict__ flat_out, float* __restrict__ out, int use_ws)
{
    __shared__ float xs[IMG];
    __shared__ float w1s[HID * NW];
    __shared__ float b1s[HID];
    __shared__ float w2s[HID];
    __shared__ float vcs[8], vss[8];
    __shared__ float meass[PADP * NW];
    __shared__ float hs[PADP * HSTRIDE];
    __shared__ float scoress[PADP];
    __shared__ float scr[256];

    const int tid  = threadIdx.x;
    const int lane = tid & 31;
    const int wave = tid >> 5;
    const int bimg = blockIdx.x;

    // ---- stage inputs -----------------------------------------------------
    for (int i = tid; i < IMG; i += 256) xs[i] = x[(size_t)bimg * IMG + i];
    w1s[tid] = w1[tid];                       // 256 == HID*NW threads/elems
    if (tid < HID) { b1s[tid] = b1[tid]; w2s[tid] = w2[tid]; }
    if (tid < 2 * NW) {
        float a = vparams[tid] * 0.5f;
        __sincosf(a, &vss[tid], &vcs[tid]);
    }
    __syncthreads();

    // ---- phase 1: 4-wire statevector sim, one patch per thread -----------
    if (tid < PATCHES) {
        const int r = tid / 14, c = tid % 14;
        float ang[4] = { xs[(2 * r) * 28 + 2 * c],     xs[(2 * r) * 28 + 2 * c + 1],
                         xs[(2 * r + 1) * 28 + 2 * c], xs[(2 * r + 1) * 28 + 2 * c + 1] };
        float cd[4], sd[4];
        #pragma unroll
        for (int w = 0; w < 4; w++) __sincosf(ang[w] * 0.5f, &sd[w], &cd[w]);

        // product state after data RYs: amp(q0..q3) = prod (q? sin : cos)
        float st[16];
        #pragma unroll
        for (int i = 0; i < 16; i++) {
            st[i] = ((i & 8) ? sd[0] : cd[0]) * ((i & 4) ? sd[1] : cd[1])
                  * ((i & 2) ? sd[2] : cd[2]) * ((i & 1) ? sd[3] : cd[3]);
        }
        // variational layers: RY(vparams) on each wire, then CNOT chain
        #pragma unroll
        for (int l = 0; l < 2; l++) {
            #pragma unroll
            for (int w = 0; w < 4; w++) {
                const int mask = 8 >> w;
                const float ct = vcs[l * 4 + w], sn = vss[l * 4 + w];
                #pragma unroll
                for (int i = 0; i < 16; i++) {
                    if (i & mask) continue;
                    const int j = i | mask;
                    const float a0 = st[i], a1 = st[j];
                    st[i] = ct * a0 - sn * a1;
                    st[j] = sn * a0 + ct * a1;
                }
            }
            #pragma unroll
            for (int g = 0; g < 3; g++) {           // CNOT(g, g+1)
                const int mc = 8 >> g, mt = 8 >> (g + 1);
                #pragma unroll
                for (int i = 0; i < 16; i++) {
                    if ((i & mc) && !(i & mt)) {
                        const int j = i | mt;
                        const float t = st[i]; st[i] = st[j]; st[j] = t;
                    }
                }
            }
        }
        // Z expectations
        float pr[16];
        #pragma unroll
        for (int i = 0; i < 16; i++) pr[i] = st[i] * st[i];
        #pragma unroll
        for (int w = 0; w < 4; w++) {
            const int mask = 8 >> w;
            float m = 0.f;
            #pragma unroll
            for (int i = 0; i < 16; i++) m += (i & mask) ? -pr[i] : pr[i];
            meass[tid * 4 + w] = m;
        }
    } else if (tid < PADP) {                       // zero-pad rows 196..207
        meass[tid * 4 + 0] = 0.f; meass[tid * 4 + 1] = 0.f;
        meass[tid * 4 + 2] = 0.f; meass[tid * 4 + 3] = 0.f;
    }
    __syncthreads();

    // ---- phase 2: h = relu(meas @ w1^T + b1) via V_WMMA_F32_16X16X4_F32 ---
    // nt = t & 3 is wave-invariant (stride 8) -> compiler hoists B and bias C.
    {
        const int halfsel = lane >> 4;             // 0: K={0,1}, 1: K={2,3}
        const int koff    = halfsel * 2;
        const int nl      = lane & 15;
        for (int t = wave; t < 13 * 4; t += 8) {   // uniform per wave
            const int mt = t >> 2, nt = t & 3;
            const int row = mt * 16 + nl;
            v2f a;  a.x  = meass[row * 4 + koff];  a.y  = meass[row * 4 + koff + 1];
            const int n = nt * 16 + nl;
            v2f bb; bb.x = w1s[n * 4 + koff];      bb.y = w1s[n * 4 + koff + 1];
            const float bias = b1s[n];
            v8f cc = { bias, bias, bias, bias, bias, bias, bias, bias };
            v8f d = __builtin_amdgcn_wmma_f32_16x16x4_f32(
                        false, a, false, bb, (short)0, cc, false, false);
            #pragma unroll
            for (int v = 0; v < 8; v++)
                hs[(mt * 16 + v + 8 * halfsel) * HSTRIDE + n] = fmaxf(d[v], 0.f);
        }
    }
    __syncthreads();

    // ---- phase 3: scores + softmax over 196 patches -----------------------
    float sv = -1e30f;
    if (tid < PATCHES) {
        float s = b2[0];
        #pragma unroll 8
        for (int h = 0; h < HID; h++) s += hs[tid * HSTRIDE + h] * w2s[h];
        scoress[tid] = s;
        sv = s;
    }
    scr[tid] = sv;
    __syncthreads();
    for (int off = 128; off > 0; off >>= 1) {
        if (tid < off) scr[tid] = fmaxf(scr[tid], scr[tid + off]);
        __syncthreads();
    }
    const float mx = scr[0];
    __syncthreads();
    const float e = (tid < PATCHES) ? __expf(sv - mx) : 0.f;
    scr[tid] = e;
    __syncthreads();
    for (int off = 128; off > 0; off >>= 1) {
        if (tid < off) scr[tid] += scr[tid + off];
        __syncthreads();
    }
    const float inv = 1.f / scr[0];
    __syncthreads();
    if (tid < PATCHES) scoress[tid] = e * inv;     // attention weights
    __syncthreads();

    // ---- phase 4: weighted flat out (or in-block logits fallback) ---------
    if (use_ws) {
        for (int i = tid; i < IMG; i += 256)
            flat_out[(size_t)bimg * IMG + i] = meass[i] * scoress[i >> 2];
    } else {
        float part = 0.f;
        if (tid < NCLS * 16) {                     // 16 threads per class
            const int o = tid >> 4, seg = tid & 15;
            const float* wr = Wf + o * IMG + seg * 49;
            for (int j = 0; j < 49; j++) {
                const int k = seg * 49 + j;
                part += meass[k] * scoress[k >> 2] * wr[j];
            }
        }
        scr[tid] = part;
        __syncthreads();
        for (int off = 8; off > 0; off >>= 1) {
            if (tid < NCLS * 16 && (tid & 15) < off) scr[tid] += scr[tid + off];
            __syncthreads();
        }
        if (tid < NCLS) scr[240 + tid] = scr[tid * 16] + bf[tid];
        __syncthreads();
        if (tid < NCLS) {
            float mmax = -1e30f;
            for (int j = 0; j < NCLS; j++) mmax = fmaxf(mmax, scr[240 + j]);
            float s = 0.f;
            for (int j = 0; j < NCLS; j++) s += __expf(scr[240 + j] - mmax);
            out[(size_t)bimg * NCLS + tid] = scr[240 + tid] - mmax - logf(s);
        }
    }
}

// ---------------------------------------------------------------------------
// Kernel 2: logits = flat @ W^T + b, fused log_softmax.
// 2 output tiles per block; 4 waves per tile split K=784 into 4x196 quarters;
// each wave interleaves TWO WMMA accumulator chains (dep depth ~25 instead of
// 196). Partials reduced through LDS, log_softmax fused in epilogue.
// ---------------------------------------------------------------------------
__global__ __launch_bounds__(256) void logits_wmma(
    const float* __restrict__ flat, const float* __restrict__ Wf,
    const float* __restrict__ bf, float* __restrict__ out,
    int Btot, int mtiles)
{
    __shared__ float part[8][16][17];
    const int lane = threadIdx.x & 31;
    const int wave = threadIdx.x >> 5;
    const int g    = wave >> 2;                    // tile slot in block (0/1)
    const int q    = wave & 3;                     // K-quarter (0..3)
    int mt = blockIdx.x * 2 + g;
    if (mt >= mtiles) mt = mtiles - 1;             // clamp (uniform per wave)

    const int halfsel = lane >> 4;
    const int koff    = halfsel * 2;
    const int nl      = lane & 15;
    int rowA = mt * 16 + nl;
    if (rowA >= Btot) rowA = Btot - 1;
    const float* arow  = flat + (size_t)rowA * IMG;
    const bool  nvalid = nl < NCLS;
    const float* brow  = Wf + (size_t)(nvalid ? nl : 0) * IMG;
    const float bias   = (nvalid && q == 0) ? bf[nl] : 0.f;

    v8f acc0 = { bias, bias, bias, bias, bias, bias, bias, bias };
    v8f acc1 = { 0.f, 0.f, 0.f, 0.f, 0.f, 0.f, 0.f, 0.f };
    const int kbeg = q * (IMG / 4);                // 196 per quarter
    const int kend = kbeg + (IMG / 4);
    for (int k0 = kbeg; k0 < kend; k0 += 8) {
        v2f a0;  a0.x = arow[k0 + koff];  a0.y = arow[k0 + koff + 1];
        v2f b0;
        b0.x = nvalid ? brow[k0 + koff]     : 0.f;
        b0.y = nvalid ? brow[k0 + koff + 1] : 0.f;
        acc0 = __builtin_amdgcn_wmma_f32_16x16x4_f32(
                   false, a0, false, b0, (short)0, acc0, false, false);
        if (k0 + 4 < kend) {                       // 196 = 24*8 + 4
            v2f a1;  a1.x = arow[k0 + 4 + koff];  a1.y = arow[k0 + 4 + koff + 1];
            v2f b1;
            b1.x = nvalid ? brow[k0 + 4 + koff]     : 0.f;
            b1.y = nvalid ? brow[k0 + 4 + koff + 1] : 0.f;
            acc1 = __builtin_amdgcn_wmma_f32_16x16x4_f32(
                       false, a1, false, b1, (short)0, acc1, false, false);
        }
    }
    acc0 = acc0 + acc1;                            // elementwise merge

    #pragma unroll
    for (int v = 0; v < 8; v++) part[wave][v + 8 * halfsel][nl] = acc0[v];
    __syncthreads();

    if (threadIdx.x < 32) {                        // 2 tiles x 16 rows
        const int gg = threadIdx.x >> 4;
        const int r  = threadIdx.x & 15;
        int mtt = blockIdx.x * 2 + gg;
        if (mtt >= mtiles) mtt = mtiles - 1;
        const int grow = mtt * 16 + r;
        if (grow < Btot) {
            float l[NCLS];
            float mmax = -1e30f;
            #pragma unroll
            for (int j = 0; j < NCLS; j++) {
                l[j] = part[gg * 4 + 0][r][j] + part[gg * 4 + 1][r][j]
                     + part[gg * 4 + 2][r][j] + part[gg * 4 + 3][r][j];
                mmax = fmaxf(mmax, l[j]);
            }
            float s = 0.f;
            #pragma unroll
            for (int j = 0; j < NCLS; j++) s += __expf(l[j] - mmax);
            const float lse = mmax + logf(s);
            #pragma unroll
            for (int j = 0; j < NCLS; j++)
                out[(size_t)grow * NCLS + j] = l[j] - mmax - lse;
        }
    }
}

// ---------------------------------------------------------------------------
extern "C" void kernel_launch(void* const* d_in, const int* in_sizes, int n_in,
                              void* d_out, int out_size, void* d_ws, size_t ws_size,
                              hipStream_t stream) {
    (void)n_in; (void)out_size;
    const float* x  = (const float*)d_in[0];
    const float* vp = (const float*)d_in[1];
    const float* w1 = (const float*)d_in[2];
    const float* b1 = (const float*)d_in[3];
    const float* w2 = (const float*)d_in[4];
    const float* b2 = (const float*)d_in[5];
    const float* Wf = (const float*)d_in[6];
    const float* bf = (const float*)d_in[7];
    float* out = (float*)d_out;

    const int Btot = in_sizes[0] / IMG;                       // 4096
    const size_t need = (size_t)Btot * IMG * sizeof(float);   // 12.8 MB
    const int use_ws = (d_ws != nullptr && ws_size >= need) ? 1 : 0;

    quanv_fused<<<Btot, 256, 0, stream>>>(x, vp, w1, b1, w2, b2, Wf, bf,
                                          (float*)d_ws, out, use_ws);
    if (use_ws) {
        const int mtiles = (Btot + 15) / 16;                  // 256
        const int blocks = (mtiles + 1) / 2;                  // 128
        logits_wmma<<<blocks, 256, 0, stream>>>((const float*)d_ws, Wf, bf,
                                                out, Btot, mtiles);
    }
}